// pmkl_loss_16638703305203
// MI455X (gfx1250) — compile-verified
//
#include <hip/hip_runtime.h>

typedef __attribute__((ext_vector_type(2))) float v2f;
typedef __attribute__((ext_vector_type(8))) float v8f;

#define TEMP_K   5.0f
#define CLIP_LO  0.005f
#define EPS_C    1.0f
#define BB       4        // batch
#define FF       8192     // feature dim

// ---------------------------------------------------------------------------
// 0) zero the two float accumulators in workspace (graph-capture safe)
// ---------------------------------------------------------------------------
__global__ void zero_ws_kernel(float* ws) {
    if (threadIdx.x < 2) ws[threadIdx.x] = 0.0f;
}

// ---------------------------------------------------------------------------
// 1) KD branch: memory-bound reduction over spatial pixels.
//    One thread per spatial index sp; loops the 4 batches so the 4 labels at
//    sp are loaded once and reused for the broadcast-count trick:
//      sum_j [label_j == pred_i]  ==  cnt_i
// ---------------------------------------------------------------------------
__global__ void kd_kernel(const float* __restrict__ te,
                          const float* __restrict__ st,
                          const int*   __restrict__ label,
                          float*       __restrict__ ws,
                          int S) {
    const int sp = blockIdx.x * blockDim.x + threadIdx.x;
    float numer = 0.0f, denom = 0.0f;

    if (sp < S) {
        int lab[BB];
#pragma unroll
        for (int j = 0; j < BB; ++j) lab[j] = label[j * S + sp];

#pragma unroll
        for (int b = 0; b < BB; ++b) {
            const int base = b * 2 * S + sp;      // [B,C=2,S] layout
            const float t0 = te[base], t1 = te[base + S];
            const float s0 = st[base], s1 = st[base + S];
            const int   lb = lab[b];

            // 2-way log-softmax CE
            const float mt   = fmaxf(t0, t1);
            const float lset = mt + __logf(__expf(t0 - mt) + __expf(t1 - mt));
            const float ms   = fmaxf(s0, s1);
            const float lses = ms + __logf(__expf(s0 - ms) + __expf(s1 - ms));
            const float te_ce = lset - (lb ? t1 : t0);
            const float st_ce = lses - (lb ? s1 : s0);
            const float mask1 = fmaxf(st_ce - te_ce, 0.0f);

            // argmax (first-max tie rule: strict >)
            const int pred = (t1 > t0) ? 1 : 0;

            // tempered softmaxes via sigmoid
            const float pte1 = 1.0f / (1.0f + __expf((t0 - t1) * (1.0f / TEMP_K)));
            const float pte0 = 1.0f - pte1;
            const float pst1 = 1.0f / (1.0f + __expf((s0 - s1) * (1.0f / TEMP_K)));
            const float pst0 = 1.0f - pst1;

            const float kd_pixel =
                -(pte0 * __logf(fmaxf(pst0, CLIP_LO)) +
                  pte1 * __logf(fmaxf(pst1, CLIP_LO)));

            int cnt = 0;
#pragma unroll
            for (int j = 0; j < BB; ++j) cnt += (lab[j] == pred) ? 1 : 0;

            const float m = mask1 * (float)cnt;
            numer += kd_pixel * m;
            denom += m;
        }
    }

    // block reduction (wave32-agnostic tree in LDS), one atomic per block
    __shared__ float sn[256];
    __shared__ float sd[256];
    const int t = threadIdx.x;
    sn[t] = numer; sd[t] = denom;
    __syncthreads();
#pragma unroll
    for (int off = 128; off > 0; off >>= 1) {
        if (t < off) { sn[t] += sn[t + off]; sd[t] += sd[t + off]; }
        __syncthreads();
    }
    if (t == 0) {
        atomicAdd(&ws[0], sn[0]);
        atomicAdd(&ws[1], sd[0]);
    }
}

// ---------------------------------------------------------------------------
// 2) Contrast branch: G = X * X^T via V_WMMA_F32_16X16X4_F32 (one wave32).
//    X rows 0..3 = teacher feats, 4..7 = student feats, 8..15 = zero.
//    A 16x4 layout: lane L holds row (L&15), K = k0 + 2*(L>>4) + {0,1};
//    B 4x16 is the mirrored striping of A^T of the same data, so A==B here.
//    Also finalizes out[0] = numer/denom from the KD workspace.
// ---------------------------------------------------------------------------
__global__ void contrast_kernel(const float* __restrict__ tf,
                                const float* __restrict__ sf,
                                const float* __restrict__ ws,
                                float*       __restrict__ out) {
    const int lane = threadIdx.x;     // 0..31, exactly one wave32, EXEC all 1s
    const int r    = lane & 15;       // matrix row index
    const int kk   = (lane >> 4) * 2; // K sub-offset within a K=4 chunk

    // rows >= 8 read valid memory but are scaled to zero (keeps EXEC full)
    const float* row = (r < BB) ? (tf + r * FF)
                    : (r < 2 * BB) ? (sf + (r - BB) * FF)
                    : tf;
    const float scale = (r < 2 * BB) ? 1.0f : 0.0f;

    v8f c = {};
    for (int k0 = 0; k0 < FF; k0 += 4) {
        v2f x;
        x.x = row[k0 + kk]     * scale;
        x.y = row[k0 + kk + 1] * scale;
        c = __builtin_amdgcn_wmma_f32_16x16x4_f32(
                /*neg_a=*/false, x, /*neg_b=*/false, x,
                /*c_mod=*/(short)0, c, /*reuse_a=*/false, /*reuse_b=*/false);
    }

    // C/D layout: VGPR v -> M = v + 8*(lane>>4), N = lane&15
    __shared__ float G[16][16];
    const int hi = lane >> 4;
#pragma unroll
    for (int v = 0; v < 8; ++v) G[v + 8 * hi][lane & 15] = c[v];
    __syncthreads();

    if (lane == 0) {
        float loss = 0.0f;
        for (int j = 0; j < BB; ++j) {
            for (int i = 0; i < BB; ++i) {
                float g = G[j][j] + G[BB + i][BB + i] - 2.0f * G[j][BB + i];
                g = fmaxf(g, 0.0f);
                const float dist = sqrtf(g) * (1.0f / (float)FF);
                if (i == j) {
                    loss += dist * dist;
                } else {
                    const float rr = fmaxf(EPS_C - dist, 0.0f);
                    loss += rr * rr;
                }
            }
        }
        out[1] = loss;            // contrast_loss
        out[0] = ws[0] / ws[1];   // kd_ce_loss (KD kernel already done: same stream)
    }
}

// ---------------------------------------------------------------------------
extern "C" void kernel_launch(void* const* d_in, const int* in_sizes, int n_in,
                              void* d_out, int out_size, void* d_ws, size_t ws_size,
                              hipStream_t stream) {
    const float* te    = (const float*)d_in[0];  // teacher_out [B,2,D,H,W]
    const float* st    = (const float*)d_in[1];  // student_out [B,2,D,H,W]
    const float* tf    = (const float*)d_in[2];  // teacher_feat [B,F]
    const float* sf    = (const float*)d_in[3];  // student_feat [B,F]
    const int*   label = (const int*)d_in[4];    // label [B,1,D,H,W]
    float* out = (float*)d_out;
    float* ws  = (float*)d_ws;

    const int S = in_sizes[4] / BB;              // D*H*W spatial size

    zero_ws_kernel<<<1, 32, 0, stream>>>(ws);

    const int blocks = (S + 255) / 256;
    kd_kernel<<<blocks, 256, 0, stream>>>(te, st, label, ws, S);

    contrast_kernel<<<1, 32, 0, stream>>>(tf, sf, ws, out);
}